// Feature2VertexLayer_38362647887989
// MI455X (gfx1250) — compile-verified
//
#include <hip/hip_runtime.h>
#include <math.h>

typedef __attribute__((ext_vector_type(2))) float v2f;
typedef __attribute__((ext_vector_type(8))) float v8f;

// Explicit global-address-space pointers: loads/stores through these must
// lower to global_load_* / global_store_* (never flat_*), so the GEMM inner
// loop only touches LOADcnt, not DScnt.
typedef const float __attribute__((address_space(1)))* gcfp;
typedef float __attribute__((address_space(1)))* gfp;

__device__ __forceinline__ float gelu_exact(float v) {
    // exact GELU: 0.5*x*(1+erf(x/sqrt(2)))
    return 0.5f * v * (1.0f + erff(v * 0.7071067811865475f));
}

__global__ void zero_f32(float* __restrict__ p, long long n) {
    long long i = (long long)blockIdx.x * blockDim.x + threadIdx.x;
    if (i < n) p[i] = 0.0f;
}

__global__ void count_deg(const int* __restrict__ dst, float* __restrict__ cnt, int E) {
    int i = blockIdx.x * blockDim.x + threadIdx.x;
    if (i < E) atomicAdd(&cnt[dst[i]], 1.0f);
}

__global__ void finalize_inv(float* __restrict__ cnt, int n) {
    int i = blockIdx.x * blockDim.x + threadIdx.x;
    if (i < n) cnt[i] = 1.0f / fmaxf(cnt[i], 1.0f);
}

// Pack W [K x Nc] into WMMA B-operand order, zero-padded to NcPad columns:
//   Wpk[((ks*2 + half)*NcPad + col)*2 + j] = W[(ks*4 + half*2 + j)*Nc + col]
// so each lane's two B values per k-step are contiguous (one b64 load).
__global__ void pack_w(const float* __restrict__ W, float* __restrict__ Wpk,
                       int K, int Nc, int NcPad) {
    int i = blockIdx.x * blockDim.x + threadIdx.x;
    if (i >= K * NcPad) return;
    int j   = i & 1;
    int q   = i >> 1;
    int col = q % NcPad;
    int qq  = q / NcPad;
    int h   = qq & 1;
    int ks  = qq >> 1;
    int row = ks * 4 + h * 2 + j;
    Wpk[i] = (col < Nc) ? W[row * Nc + col] : 0.0f;
}

// Scatter-add yl[src] rows into agg[dst] rows. VEC=4 when dout%4==0 (float4
// gather, 4 atomics), VEC=1 otherwise (last layer, dout=3).
template<int VEC>
__global__ void scatter_add(const float* __restrict__ yl, const int* __restrict__ src,
                            const int* __restrict__ dst, float* __restrict__ agg,
                            int E, int dout) {
    const int chunks = dout / VEC;
    long long tid = (long long)blockIdx.x * blockDim.x + threadIdx.x;
    if (tid >= (long long)E * chunks) return;
    int e = (int)(tid / chunks);
    int c = (int)(tid % chunks);
    int s = src[e], d = dst[e];
    if (VEC == 4) {
        const float4 v = *(const float4*)(yl + (size_t)s * dout + c * 4);
        float* p = agg + (size_t)d * dout + c * 4;
        atomicAdd(p + 0, v.x);
        atomicAdd(p + 1, v.y);
        atomicAdd(p + 2, v.z);
        atomicAdd(p + 3, v.w);
    } else {
        atomicAdd(agg + (size_t)d * dout + c, yl[(size_t)s * dout + c]);
    }
}

// agg[i,f] = agg[i,f] * inv_deg[i] + bias[f]   (in place)
__global__ void combine(float* __restrict__ agg, const float* __restrict__ inv,
                        const float* __restrict__ bias, int n, int dout) {
    long long i = (long long)blockIdx.x * blockDim.x + threadIdx.x;
    if (i >= (long long)n * dout) return;
    int node = (int)(i / dout);
    int f = (int)(i % dout);
    agg[i] = agg[i] * inv[node] + bias[f];
}

// D = [gelu]( (ACC ? Cin : 0) + A @ W )
// A: [M x K] row-major (M % 16 == 0, K % 4 == 0), Wpk: packed/padded B operand
// (see pack_w).  Each wave computes MT vertically-stacked 16x16 output tiles at
// one N-tile: per k-step it issues MT A b64 loads + 1 B b64 load into distinct
// locals (load phase), then runs MT independent V_WMMA_F32_16X16X4_F32 ops
// (compute phase), so loads overlap the matrix pipe.  Pointers are explicit
// address_space(1) so loads stay global (LOADcnt only) while remaining real
// pointers: the per-k advances fold into immediate offsets and each adjacent
// pair merges into one global_load_b64.  Inner loop has no masking; only the
// final store is masked (col < Nc, valid tile).
// VGPR layouts per CDNA5 ISA 7.12.2:
//   A (16x4 f32): lanes 0-15 -> M=l, K = k+0 (v0), k+1 (v1);
//                 lanes 16-31 -> M=l, K = k+2 (v0), k+3 (v1)
//   B (4x16 f32): v0: lanes 0-15 = row k+0, lanes 16-31 = row k+2; v1: k+1/k+3
//   C/D (16x16):  VGPR g: lanes 0-15 -> M=g, lanes 16-31 -> M=g+8, N=lane&15
template<bool ACC, bool GELU, int MT>
__global__ __launch_bounds__(256) void gemm_wmma(
        const float* __restrict__ A, const float* __restrict__ Wpk,
        const float* __restrict__ Cin, float* __restrict__ D,
        int K, int Nc, int NcPad, int tilesM, int tilesN, int totalTiles) {
    int wave = threadIdx.x >> 5;
    int tile = blockIdx.x * (blockDim.x >> 5) + wave;
    if (tile >= totalTiles) return;   // wave-uniform: EXEC all-ones past here
    int group = tile / tilesN;        // vertical group of MT M-tiles
    int tn    = tile - group * tilesN;
    int tmBase = group * MT;
    int lane = threadIdx.x & 31;
    int half = lane >> 4;             // 0: lanes 0-15, 1: lanes 16-31
    int l = lane & 15;
    int col = tn * 16 + l;            // always < NcPad

    // Packed-B pointer: lane's two values adjacent; advance 1 k-step (4 rows).
    gcfp Wb = (gcfp)Wpk + ((size_t)half * NcPad + col) * 2;
    const size_t wstep = (size_t)4 * NcPad;   // floats per k-step

    // A row pointers (row-clamped so remainder groups never read OOB).
    gcfp Ap[MT];
    #pragma unroll
    for (int t = 0; t < MT; ++t) {
        int rt = tmBase + t;
        if (rt > tilesM - 1) rt = tilesM - 1;
        Ap[t] = (gcfp)A + (size_t)(rt * 16 + l) * K + half * 2;
    }

    v8f acc[MT];
    #pragma unroll
    for (int t = 0; t < MT; ++t) acc[t] = (v8f){};

    for (int k = 0; k < K; k += 4) {
        // ---- load phase: distinct locals -> batched global_load_b64
        float bx = Wb[0];
        float by = Wb[1];
        Wb += wstep;
        float ax[MT], ay[MT];
        #pragma unroll
        for (int t = 0; t < MT; ++t) {
            ax[t] = Ap[t][0];
            ay[t] = Ap[t][1];
            Ap[t] += 4;
        }
        // ---- compute phase: MT independent WMMAs on the matrix pipe
        v2f b;
        b.x = bx;
        b.y = by;
        #pragma unroll
        for (int t = 0; t < MT; ++t) {
            v2f a;
            a.x = ax[t];
            a.y = ay[t];
            acc[t] = __builtin_amdgcn_wmma_f32_16x16x4_f32(
                         false, a, false, b, (short)0, acc[t], false, false);
        }
    }

    if (col < Nc) {
        gcfp Cg = (gcfp)Cin;
        gfp  Dg = (gfp)D;
        #pragma unroll
        for (int t = 0; t < MT; ++t) {
            int rt = tmBase + t;
            if (rt < tilesM) {                 // wave-uniform guard
                #pragma unroll
                for (int g = 0; g < 8; ++g) {
                    int row = rt * 16 + half * 8 + g;
                    size_t off = (size_t)row * Nc + col;
                    float v = acc[t][g];
                    if (ACC) v += Cg[off];
                    if (GELU) v = gelu_exact(v);
                    Dg[off] = v;
                }
            }
        }
    }
}

extern "C" void kernel_launch(void* const* d_in, const int* in_sizes, int n_in,
                              void* d_out, int out_size, void* d_ws, size_t ws_size,
                              hipStream_t stream) {
    static const int DIMS[5] = {256, 192, 128, 64, 3};
    constexpr int MT = 5;   // M-tiles per wave (3125 M-tiles = 5^5 -> exact)

    const float* feat  = (const float*)d_in[0];
    const int*   edges = (const int*)d_in[1];
    const int N = in_sizes[0] / DIMS[0];
    const int E = in_sizes[1] / 2;
    const int* src = edges;
    const int* dst = edges + E;

    const float* Wl[4] = {(const float*)d_in[2], (const float*)d_in[5],
                          (const float*)d_in[8], (const float*)d_in[11]};
    const float* Wr[4] = {(const float*)d_in[3], (const float*)d_in[6],
                          (const float*)d_in[9], (const float*)d_in[12]};
    const float* Bb[4] = {(const float*)d_in[4], (const float*)d_in[7],
                          (const float*)d_in[10], (const float*)d_in[13]};

    float* ws    = (float*)d_ws;
    float* inv   = ws;                                  // [N]
    float* bufP  = ws + N;                              // [N x 192]
    float* bufQ  = bufP + (size_t)N * 192;              // [N x 192] (agg)
    float* bufR  = bufQ + (size_t)N * 192;              // [N x 192]
    float* wPkL  = bufR + (size_t)N * 192;              // [256 x 208] packed Wl
    float* wPkR  = wPkL + 256 * 208;                    // [256 x 208] packed Wr

    // --- degree reciprocals (once; edges are fixed across layers) ---
    zero_f32<<<(N + 255) / 256, 256, 0, stream>>>(inv, N);
    count_deg<<<(E + 255) / 256, 256, 0, stream>>>(dst, inv, E);
    finalize_inv<<<(N + 255) / 256, 256, 0, stream>>>(inv, N);

    const float* x = feat;
    for (int L = 0; L < 4; ++L) {
        const int K    = DIMS[L];
        const int dout = DIMS[L + 1];
        float* yl   = (L & 1) ? bufR : bufP;
        float* agg  = bufQ;
        float* xout = (L == 3) ? (float*)d_out : yl;    // yl dead once scatter done

        const int tilesM = N / 16;                      // 50000 / 16 = 3125 exact
        const int tilesN = (dout + 15) / 16;
        const int NcPad  = tilesN * 16;
        const int groups = (tilesM + MT - 1) / MT;
        const int total  = groups * tilesN;
        const int blocks = (total + 7) / 8;             // 8 waves / 256-thread block

        // Pack both weight matrices into WMMA operand order (cheap: <=49K elts).
        const int padN = K * NcPad;
        pack_w<<<(padN + 255) / 256, 256, 0, stream>>>(Wl[L], wPkL, K, dout, NcPad);
        pack_w<<<(padN + 255) / 256, 256, 0, stream>>>(Wr[L], wPkR, K, dout, NcPad);

        // yl = x @ Wl   (aggregate AFTER the GEMM: dout < din every layer)
        gemm_wmma<false, false, MT><<<blocks, 256, 0, stream>>>(
            x, wPkL, nullptr, yl, K, dout, NcPad, tilesM, tilesN, total);

        const long long na = (long long)N * dout;
        zero_f32<<<(unsigned)((na + 255) / 256), 256, 0, stream>>>(agg, na);

        if (dout % 4 == 0) {
            long long work = (long long)E * (dout / 4);
            scatter_add<4><<<(unsigned)((work + 255) / 256), 256, 0, stream>>>(
                yl, src, dst, agg, E, dout);
        } else {
            long long work = (long long)E * dout;
            scatter_add<1><<<(unsigned)((work + 255) / 256), 256, 0, stream>>>(
                yl, src, dst, agg, E, dout);
        }

        // agg = agg * inv_deg + b
        combine<<<(unsigned)((na + 255) / 256), 256, 0, stream>>>(
            agg, inv, Bb[L], N, dout);

        // xout = [gelu]( agg + x @ Wr )  (activation fused into GEMM epilogue)
        if (L < 3) {
            gemm_wmma<true, true, MT><<<blocks, 256, 0, stream>>>(
                x, wPkR, agg, xout, K, dout, NcPad, tilesM, tilesN, total);
        } else {
            gemm_wmma<true, false, MT><<<blocks, 256, 0, stream>>>(
                x, wPkR, agg, xout, K, dout, NcPad, tilesM, tilesN, total);
        }
        x = xout;
    }
}